// EarlyExitGateLoss_75265006895580
// MI455X (gfx1250) — compile-verified
//
#include <hip/hip_runtime.h>
#include <stdint.h>

#define ALPHA 0.5f
#define WAVE 32

// ---------------------------------------------------------------------------
// CDNA5 async global->LDS helpers (inline asm; portable across toolchains).
// vdst = 32-bit LDS byte offset VGPR, vaddr = 64-bit global address VGPR pair.
// Non-temporal: single-use 524 MB stream, keep it out of L2's way.
// ---------------------------------------------------------------------------
__device__ __forceinline__ void async_load_f4(const void* gsrc, uint32_t lds_off) {
  uint64_t ga = (uint64_t)(uintptr_t)gsrc;
  asm volatile("global_load_async_to_lds_b128 %0, %1, off th:TH_LOAD_NT"
               :: "v"(lds_off), "v"(ga) : "memory");
}
#define WAIT_ASYNC(n) asm volatile("s_wait_asynccnt " #n ::: "memory")

// Online softmax accumulator update: one v_exp_f32 per element.
__device__ __forceinline__ void online_upd(float x, float& m, float& s) {
  float d = x - m;                    // +inf on first element (m = -inf)
  if (d > 0.0f) {
    s = __builtin_fmaf(s, __expf(-d), 1.0f);
    m = x;
  } else {
    s += __expf(d);
  }
}

// Merge two (max, sumexp) pairs.
__device__ __forceinline__ void combine(float& m, float& s, float mo, float so) {
  float nm = fmaxf(m, mo);
  s = s * __expf(m - nm) + so * __expf(mo - nm);
  m = nm;
}

// ---------------------------------------------------------------------------
// Kernel 1: per-(b,k) row cross-entropy via streaming log-sum-exp.
// grid = (K, B): no integer division. One 256-thread block (8 wave32) per row.
// 4-deep async ring buffer; each lane owns a private 16 B slice per stage, so
// only per-wave s_wait_asynccnt is needed (no block barriers in the hot loop).
// ---------------------------------------------------------------------------
__global__ __launch_bounds__(256) void ee_ce_kernel(
    const float* __restrict__ y_hats,  // [B*K*C]
    const int*   __restrict__ ys,      // [B]
    float*       __restrict__ ce,      // [B*K] (workspace)
    int C) {
  const int k   = blockIdx.x;
  const int b   = blockIdx.y;
  const int K   = gridDim.x;
  const int r   = b * K + k;
  const int tid = threadIdx.x;

  const float*  row  = y_hats + (size_t)r * (size_t)C;
  const float4* rowv = (const float4*)row;

  __shared__ float4 abuf[4][256];      // 16 KB ring buffer, 16 B per lane/stage
  __shared__ float  mred[8];
  __shared__ float  sred[8];

  // Label logit (single scalar load, latency hidden under the stream).
  float xlab = 0.0f;
  if (tid == 0) xlab = row[ys[b]];

  const int nV4     = C >> 2;          // float4 count (C % 4 handled below)
  const int nStages = nV4 >> 8;        // full 256-float4 stages
  const int tail    = nV4 & 255;

  const uint32_t lds_base = (uint32_t)(uintptr_t)&abuf[0][tid];
  // abuf[j][256] stride in bytes = 256*16 = 4096 -> slot j at lds_base + (j<<12)

  // Two independent accumulator chains for exp-latency ILP.
  float m0 = -INFINITY, s0 = 0.0f;
  float m1 = -INFINITY, s1 = 0.0f;

  if (nStages > 0) {
    const int pre = nStages < 3 ? nStages : 3;
    for (int i = 0; i < pre; ++i)
      async_load_f4(rowv + (size_t)i * 256 + tid,
                    lds_base + ((uint32_t)(i & 3) << 12));

    for (int st = 0; st < nStages; ++st) {
      if (st + 3 < nStages) {          // keep 3 stages in flight behind us
        async_load_f4(rowv + (size_t)(st + 3) * 256 + tid,
                      lds_base + ((uint32_t)((st + 3) & 3) << 12));
        WAIT_ASYNC(0x3);
      } else if (st + 2 < nStages) {
        WAIT_ASYNC(0x2);
      } else if (st + 1 < nStages) {
        WAIT_ASYNC(0x1);
      } else {
        WAIT_ASYNC(0x0);
      }
      float4 v = abuf[st & 3][tid];    // each lane reads only its own bytes
      online_upd(v.x, m0, s0);
      online_upd(v.y, m1, s1);
      online_upd(v.z, m0, s0);
      online_upd(v.w, m1, s1);
    }
  }
  if (tid < tail) {                    // residual float4s: direct load
    float4 v = rowv[(size_t)nStages * 256 + tid];
    online_upd(v.x, m0, s0);
    online_upd(v.y, m1, s1);
    online_upd(v.z, m0, s0);
    online_upd(v.w, m1, s1);
  }
  if (tid == 0) {                      // scalar remainder if C % 4 != 0
    for (int i = nV4 << 2; i < C; ++i) online_upd(row[i], m0, s0);
  }

  // Fold the two chains, then wave32 reduction of (m, s).
  combine(m0, s0, m1, s1);
  #pragma unroll
  for (int off = 16; off > 0; off >>= 1) {
    float mo = __shfl_xor(m0, off, WAVE);
    float so = __shfl_xor(s0, off, WAVE);
    combine(m0, s0, mo, so);
  }
  const int wid = tid >> 5, lane = tid & 31;
  if (lane == 0) { mred[wid] = m0; sred[wid] = s0; }
  __syncthreads();
  if (tid == 0) {
    float M = mred[0], S = sred[0];
    #pragma unroll
    for (int w = 1; w < 8; ++w) combine(M, S, mred[w], sred[w]);
    // ce = -(x_label - (M + log S))
    ce[r] = (M + __logf(S)) - xlab;
  }
}

// ---------------------------------------------------------------------------
// Kernel 2: gating recurrence over [B, K] + final reduction to 3 scalars.
// ---------------------------------------------------------------------------
__global__ __launch_bounds__(512) void ee_gate_kernel(
    const float* __restrict__ ce,     // [B*K]
    const float* __restrict__ g,      // [B*(K-1)]
    const float* __restrict__ costs,  // [K]
    float*       __restrict__ out,    // [3]: loss, term_gate, term_cost
    int B, int K) {
  const int tid = threadIdx.x;
  float gacc = 0.0f, cacc = 0.0f;

  for (int b = tid; b < B; b += blockDim.x) {
    const float* gb = g  + (size_t)b * (size_t)(K - 1);
    const float* cb = ce + (size_t)b * (size_t)K;
    float excl = 1.0f;                          // prod of (1-g) so far
    for (int kk = 0; kk < K - 1; ++kk) {
      float gk = gb[kk];
      float w  = excl * gk;
      gacc = __builtin_fmaf(w, cb[kk],    gacc);
      cacc = __builtin_fmaf(w, costs[kk], cacc);
      excl *= (1.0f - gk);
    }
    gacc = __builtin_fmaf(excl, cb[K - 1],    gacc);  // w[K-1] = cumprod tail
    cacc = __builtin_fmaf(excl, costs[K - 1], cacc);
  }

  #pragma unroll
  for (int off = 16; off > 0; off >>= 1) {
    gacc += __shfl_xor(gacc, off, WAVE);
    cacc += __shfl_xor(cacc, off, WAVE);
  }
  __shared__ float gs[16], cs[16];
  const int wid = tid >> 5, lane = tid & 31;
  if (lane == 0) { gs[wid] = gacc; cs[wid] = cacc; }
  __syncthreads();
  if (tid == 0) {
    float G = 0.0f, Cc = 0.0f;
    const int nw = (int)(blockDim.x >> 5);
    for (int w = 0; w < nw; ++w) { G += gs[w]; Cc += cs[w]; }
    const float invB = 1.0f / (float)B;
    const float tg = (1.0f - ALPHA) * (G * invB);
    const float tc = ALPHA * (Cc * invB);
    out[0] = tg + tc;
    out[1] = tg;
    out[2] = tc;
  }
}

// ---------------------------------------------------------------------------
// Launch
// ---------------------------------------------------------------------------
extern "C" void kernel_launch(void* const* d_in, const int* in_sizes, int n_in,
                              void* d_out, int out_size, void* d_ws, size_t ws_size,
                              hipStream_t stream) {
  const int*   ys    = (const int*)d_in[0];    // [B]
  const float* yh    = (const float*)d_in[1];  // [B*K*C]
  const float* g     = (const float*)d_in[2];  // [B*(K-1)]
  const float* costs = (const float*)d_in[3];  // [K]
  float* out = (float*)d_out;

  const int B = in_sizes[0];
  const int K = in_sizes[3];
  const int C = (int)((long long)in_sizes[1] / ((long long)B * (long long)K));

  float* ce = (float*)d_ws;                    // B*K floats of scratch

  dim3 grid1(K, B);
  ee_ce_kernel<<<grid1, 256, 0, stream>>>(yh, ys, ce, C);
  ee_gate_kernel<<<1, 512, 0, stream>>>(ce, g, costs, out, B, K);
}